// MoERouter_19396072309350
// MI455X (gfx1250) — compile-verified
//
#include <hip/hip_runtime.h>
#include <hip/hip_bf16.h>

typedef float v2f __attribute__((ext_vector_type(2)));
typedef float v8f __attribute__((ext_vector_type(8)));
typedef int   v4i __attribute__((ext_vector_type(4)));

typedef __attribute__((address_space(1))) v4i global_v4i;   // HBM side
typedef __attribute__((address_space(3))) v4i lds_v4i;      // LDS side

#define D_DIM   2048
#define E_DIM   8
#define EPAD    16
#define TOKENS  16384
#define TPB     128      // tokens per block
#define DCHUNK  64       // K depth staged per iteration
#define XSTR    68       // row stride (words): 64 + 4 pad -> conflict-free b64 frag reads
#define WSTR    68
#define XBUF    (TPB * XSTR)
#define WBUF    (EPAD * WSTR)

#if __has_builtin(__builtin_amdgcn_global_load_async_to_lds_b128)
#define HAVE_ASYNC_LDS 1
#else
#define HAVE_ASYNC_LDS 0
#endif

__device__ __forceinline__ void wait_async_done() {
#if __has_builtin(__builtin_amdgcn_s_wait_asynccnt)
    __builtin_amdgcn_s_wait_asynccnt(0);
#elif HAVE_ASYNC_LDS
    asm volatile("s_wait_asynccnt 0x0" ::: "memory");
#endif
}

__global__ __launch_bounds__(128) void moe_router_wmma(
    const float* __restrict__ x,
    const float* __restrict__ w,
    int*   __restrict__ out_idx,
    float* __restrict__ out_gates,
    float* __restrict__ out_logits)
{
    __shared__ float sX[2 * XBUF];       // double-buffered [token][k] tiles
    __shared__ float sW[2 * WBUF];       // double-buffered [expert][k], padded 8 -> 16
    __shared__ float sL[TPB * E_DIM];    // logits scratch for top-k phase

    const int tid      = threadIdx.x;    // 128 threads = 4 waves
    const int wave     = tid >> 5;
    const int lane     = tid & 31;
    const int half     = lane >> 4;      // lane half selects K sub-pair (32-bit A/B layout)
    const int l16      = lane & 15;
    const int blockTok = blockIdx.x * TPB;
    const int waveTok  = wave * 32;      // 32-token tile per wave (dual accumulators)

    // ---- issue one chunk's fill of buffer b (async HBM->LDS when available) ----
    auto stage = [&](int b, int kb) {
        float* sXb = sX + b * XBUF;
        float* sWb = sW + b * WBUF;
        for (int q = tid; q < (TPB * DCHUNK / 4); q += 128) {
            const int tLocal = q >> 4;                 // 16 float4 per token-chunk
            const int dq     = (q & 15) << 2;
            const float* g = x + (size_t)(blockTok + tLocal) * D_DIM + kb + dq;
            float* l = &sXb[tLocal * XSTR + dq];
#if HAVE_ASYNC_LDS
            __builtin_amdgcn_global_load_async_to_lds_b128(
                (global_v4i*)g, (lds_v4i*)l, 0, 0);
#else
            *(float4*)l = *(const float4*)g;
#endif
        }
        {   // W chunk: 8 experts x 16 float4 = 128 -> exactly one per thread
            const int e  = tid >> 4;
            const int dq = (tid & 15) << 2;
            const float* g = w + e * D_DIM + kb + dq;
            float* l = &sWb[e * WSTR + dq];
#if HAVE_ASYNC_LDS
            __builtin_amdgcn_global_load_async_to_lds_b128(
                (global_v4i*)g, (lds_v4i*)l, 0, 0);
#else
            *(float4*)l = *(const float4*)g;
#endif
        }
    };

    // ---- zero the expert-pad rows (8..15) of both W buffers, once ----
    for (int i = tid; i < 2 * 8 * WSTR; i += 128) {
        const int b = i / (8 * WSTR);
        const int r = i % (8 * WSTR);
        sW[b * WBUF + 8 * WSTR + r] = 0.0f;
    }

    v8f acc0 = {};
    v8f acc1 = {};

    // ---- prologue: fill buffer 0 ----
    stage(0, 0);
    wait_async_done();
    __syncthreads();

    int buf = 0;
    for (int kb = 0; kb < D_DIM; kb += DCHUNK) {
        // overlap: async-fill the other buffer while WMMAs chew on this one
        if (kb + DCHUNK < D_DIM) stage(buf ^ 1, kb + DCHUNK);

        const float* sXb = sX + buf * XBUF;
        const float* sWb = sW + buf * WBUF;

        #pragma unroll
        for (int kk = 0; kk < DCHUNK; kk += 4) {
            const int kA = kk + 2 * half;              // lane half -> K={0,1} vs K={2,3}
            const v2f a0 = *(const v2f*)&sXb[(waveTok +      l16) * XSTR + kA];
            const v2f a1 = *(const v2f*)&sXb[(waveTok + 16 + l16) * XSTR + kA];
            const v2f b  = *(const v2f*)&sWb[l16 * WSTR + kA];
            acc0 = __builtin_amdgcn_wmma_f32_16x16x4_f32(false, a0, false, b,
                                                         (short)0, acc0, false, false);
            acc1 = __builtin_amdgcn_wmma_f32_16x16x4_f32(false, a1, false, b,
                                                         (short)0, acc1, false, false);
        }

        wait_async_done();       // next chunk's fill complete (DS covered by barrier)
        __syncthreads();         // publish new buffer / release old one
        buf ^= 1;
    }

    // ---- scatter logits: VGPR v -> token (+8 per lane half), lane&15 -> expert ----
    if (l16 < E_DIM) {
        #pragma unroll
        for (int v = 0; v < 8; ++v) {
            const int t0 = waveTok + v + 8 * half;
            const int t1 = t0 + 16;
            const float v0 = acc0[v];
            const float v1 = acc1[v];
            sL[t0 * E_DIM + l16] = v0;
            sL[t1 * E_DIM + l16] = v1;
            out_logits[(size_t)(blockTok + t0) * E_DIM + l16] = v0;
            out_logits[(size_t)(blockTok + t1) * E_DIM + l16] = v1;
        }
    }
    __syncthreads();

    // ---- top-2 + softmax, one thread per token (strict > = lower-index tie-break) ----
    {
        float best0 = -__builtin_inff(), best1 = -__builtin_inff();
        int i0 = 0, i1 = 0;
        #pragma unroll
        for (int e = 0; e < E_DIM; ++e) {
            const float v = sL[tid * E_DIM + e];
            if (v > best0)      { best1 = best0; i1 = i0; best0 = v; i0 = e; }
            else if (v > best1) { best1 = v; i1 = e; }
        }
        const float e1  = __expf(best1 - best0);
        const float inv = 1.0f / (1.0f + e1);
        const int tok = blockTok + tid;
        out_idx[tok * 2 + 0]   = i0;
        out_idx[tok * 2 + 1]   = i1;
        out_gates[tok * 2 + 0] = inv;
        out_gates[tok * 2 + 1] = e1 * inv;
    }
}

extern "C" void kernel_launch(void* const* d_in, const int* in_sizes, int n_in,
                              void* d_out, int out_size, void* d_ws, size_t ws_size,
                              hipStream_t stream) {
    const float* x = (const float*)d_in[0];          // [4,4096,2048] fp32
    const float* w = (const float*)d_in[1];          // [8,2048] fp32
    float* out = (float*)d_out;
    int*   out_idx    = (int*)out;                   // [4,4096,2] int32 (bit-cast region)
    float* out_gates  = out + TOKENS * 2;            // [4,4096,2] fp32
    float* out_logits = out + TOKENS * 4;            // [4,4096,8] fp32

    dim3 grid(TOKENS / TPB);
    dim3 block(128);
    moe_router_wmma<<<grid, block, 0, stream>>>(x, w, out_idx, out_gates, out_logits);
}